// FSQQuantizer_36524401885603
// MI455X (gfx1250) — compile-verified
//
#include <hip/hip_runtime.h>
#include <cstdint>

// FSQ quantizer, MI455X (gfx1250).
// Memory-bound: ~266MB traffic -> ~11.4us floor at 23.3 TB/s.
// CDNA5 paths: async global->LDS loads (ASYNCcnt, 4-deep per-wave ring) with
// constant-immediate waits in the steady loop, native v_tanh_f32, NT stores.

#define THREADS     256
#define MAIN_BLOCKS 4096
#define NCODES      4096        // 8^4 mixed codes
#define DEPTH       4           // async pipeline depth
#define BUFBYTES    (THREADS * 16)
constexpr float BETA = 0.25f;

typedef __attribute__((ext_vector_type(4))) float v4f;

__device__ __forceinline__ uint32_t lds_off(const void* p) {
    // Low 32 bits of a flat shared-aperture address == LDS byte offset (ISA 10.2).
    return (uint32_t)(uintptr_t)p;
}

__device__ __forceinline__ void async_load16(uint32_t ldsAddr, uint32_t byteOff,
                                             unsigned long long base) {
    asm volatile("global_load_async_to_lds_b128 %0, %1, %2 offset:0"
                 :: "v"(ldsAddr), "v"(byteOff), "s"(base) : "memory");
}

__device__ __forceinline__ void wait_async_le(int n) {  // drain path only (<=3 uses)
    switch (n) {
    case 0:  asm volatile("s_wait_asynccnt 0x0" ::: "memory"); break;
    case 1:  asm volatile("s_wait_asynccnt 0x1" ::: "memory"); break;
    default: asm volatile("s_wait_asynccnt 0x2" ::: "memory"); break;
    }
}

__device__ __forceinline__ float tanh_fast(float x) {
    float r;
    asm("v_tanh_f32 %0, %1" : "=v"(r) : "v"(x));   // native CDNA5 trans op
    return r;
}

__global__ __launch_bounds__(THREADS) void fsq_init(uint32_t* occ, float* lossAcc) {
    int t = blockIdx.x * blockDim.x + threadIdx.x;
    for (int i = t; i < NCODES; i += gridDim.x * blockDim.x) occ[i] = 0u;
    if (t == 0) *lossAcc = 0.0f;
}

__global__ __launch_bounds__(THREADS) void fsq_main(
        const float* __restrict__ z_e,     // [N*32] f32
        const float* __restrict__ grid,    // [8] f32 (uniform)
        float* __restrict__ out_q,         // [N*32] f32
        float* __restrict__ out_idx,       // [N] f32 (mixed code)
        uint32_t* __restrict__ occ,        // [4096]
        float* __restrict__ lossAcc,
        int nChunks)                       // (N*32)/4 float4 chunks
{
    __shared__ v4f   buf[DEPTH][THREADS];  // per-wave ring (each lane owns its slot)
    __shared__ float red[THREADS];

    const int tid    = threadIdx.x;
    const int stride = gridDim.x * blockDim.x;
    const int c0     = blockIdx.x * blockDim.x + tid;

    // Uniform-grid quantizer parameters (uniform address -> scalar loads).
    const float g0   = grid[0];
    const float g7   = grid[7];
    const float step = (g7 - g0) * (1.0f / 7.0f);
    const float inv  = 7.0f / (g7 - g0);

    const uint32_t ldsBase = lds_off(&buf[0][0]) + (uint32_t)tid * 16u;
    const unsigned long long base = (unsigned long long)(uintptr_t)z_e;

    float lsum = 0.0f;

    auto process = [&](int cc, v4f z) {
        float zn[4] = { z.x, z.y, z.z, z.w };
        float q[4];
        int   idx[4];
#pragma unroll
        for (int k = 0; k < 4; ++k) {
            float t  = tanh_fast(zn[k]);            // v_tanh_f32
            float fi = rintf((t - g0) * inv);       // uniform-grid argmin
            fi       = fminf(fmaxf(fi, 0.0f), 7.0f);
            idx[k]   = (int)fi;
            float zq = fmaf(fi, step, g0);
            float d  = zq - t;
            lsum     = fmaf(d, d, lsum);
            q[k]     = zq;
        }
        v4f qv = { q[0], q[1], q[2], q[3] };
        __builtin_nontemporal_store(qv, (v4f*)out_q + cc);   // b128 store, TH=NT
        if ((cc & 7) == 0) {            // this lane holds dims 0..3 of a row
            int row   = cc >> 3;
            int mixed = idx[0] + (idx[1] << 3) + (idx[2] << 6) + (idx[3] << 9);
            __builtin_nontemporal_store((float)mixed, out_idx + row);
            occ[mixed] = 1u;            // benign race (all write 1)
        }
    };

    // Per-thread iteration count; steady phase keeps DEPTH loads in flight.
    const int iters = (c0 < nChunks) ? ((nChunks - 1 - c0) / stride + 1) : 0;
    const int full  = (iters > (DEPTH - 1)) ? (iters - (DEPTH - 1)) : 0;

    // Prologue: fill DEPTH-1 pipeline stages.
#pragma unroll
    for (int d = 0; d < DEPTH - 1; ++d) {
        if (d < iters)
            async_load16(ldsBase + (uint32_t)d * BUFBYTES,
                         (uint32_t)(c0 + d * stride) * 16u, base);
    }

    // Steady state: constant-immediate waits, unconditional refill.
    int c = c0;
    for (int i = 0; i < full; ++i, c += stride) {
        asm volatile("s_wait_dscnt 0x0" ::: "memory");   // reused slot's LDS read retired
        async_load16(ldsBase + (uint32_t)(((i + DEPTH - 1) & (DEPTH - 1)) * BUFBYTES),
                     (uint32_t)(c + (DEPTH - 1) * stride) * 16u, base);
        asm volatile("s_wait_asynccnt 0x3" ::: "memory"); // in-order -> oldest ready
        process(c, buf[i & (DEPTH - 1)][tid]);            // ds_load_b128
    }

    // Drain: <= DEPTH-1 iterations, descending wait levels.
#pragma unroll
    for (int d = 0; d < DEPTH - 1; ++d) {
        int i = full + d;
        if (i >= iters) break;
        wait_async_le(iters - i - 1);
        process(c, buf[i & (DEPTH - 1)][tid]);
        c += stride;
    }

    // Block loss reduction -> single atomic per block.
    red[tid] = lsum;
    __syncthreads();
#pragma unroll
    for (int s = THREADS / 2; s > 0; s >>= 1) {
        if (tid < s) red[tid] += red[tid + s];
        __syncthreads();
    }
    if (tid == 0) atomicAdd(lossAcc, red[0]);
}

__global__ __launch_bounds__(THREADS) void fsq_final(
        const uint32_t* __restrict__ occ,
        const float* __restrict__ lossAcc,
        float* __restrict__ out_loss,
        float* __restrict__ out_perp,
        int nElems, int nRows)
{
    __shared__ int red[THREADS];
    int t = threadIdx.x;
    int cnt = 0;
    for (int i = t; i < NCODES; i += THREADS) cnt += (occ[i] != 0u) ? 1 : 0;
    red[t] = cnt;
    __syncthreads();
#pragma unroll
    for (int s = THREADS / 2; s > 0; s >>= 1) {
        if (t < s) red[t] += red[t + s];
        __syncthreads();
    }
    if (t == 0) {
        float sum = *lossAcc;
        *out_loss = (1.0f + BETA) * sum / (float)nElems;   // codebook + beta*commitment
        *out_perp = (float)red[0] / (float)nRows;
    }
}

extern "C" void kernel_launch(void* const* d_in, const int* in_sizes, int n_in,
                              void* d_out, int out_size, void* d_ws, size_t ws_size,
                              hipStream_t stream) {
    const float* z_e  = (const float*)d_in[0];   // [N*32] f32
    const float* grid = (const float*)d_in[1];   // [8] f32

    const int nElems  = in_sizes[0];             // N*32
    const int nChunks = nElems / 4;              // float4 chunks
    const int nRows   = nElems / 32;             // N

    float* out      = (float*)d_out;
    float* out_q    = out;                       // [N*32]
    float* out_idx  = out + nElems;              // [N] mixed indices (as f32)
    float* out_loss = out + nElems + nRows;      // scalar
    float* out_perp = out_loss + 1;              // scalar

    uint32_t* occ    = (uint32_t*)d_ws;                          // 4096 * u32
    float*    lossAc = (float*)((char*)d_ws + NCODES * sizeof(uint32_t));

    fsq_init <<<16,          THREADS, 0, stream>>>(occ, lossAc);
    fsq_main <<<MAIN_BLOCKS, THREADS, 0, stream>>>(z_e, grid, out_q, out_idx,
                                                   occ, lossAc, nChunks);
    fsq_final<<<1,           THREADS, 0, stream>>>(occ, lossAc, out_loss, out_perp,
                                                   nElems, nRows);
    (void)n_in; (void)out_size; (void)ws_size;
}